// MultiScaleLocalPatchesTorch_33174327394663
// MI455X (gfx1250) — compile-verified
//
#include <hip/hip_runtime.h>

typedef float v2f __attribute__((ext_vector_type(2)));
typedef float v8f __attribute__((ext_vector_type(8)));

#define N_PTS 8192
#define TOPK  33
#define CHUNK 512
#define NTILES_PER_CHUNK (CHUNK / 16)   // 32 tiles of 16 columns
#define NCHUNKS (N_PTS / CHUNK)         // 16 chunks

// ---------------------------------------------------------------------------
// Kernel 1: squared norms of the 3-D points
// ---------------------------------------------------------------------------
__global__ void sq_kernel(const float* __restrict__ pts, float* __restrict__ sq) {
    int i = blockIdx.x * blockDim.x + threadIdx.x;
    if (i < N_PTS) {
        float x = pts[3 * i + 0];
        float y = pts[3 * i + 1];
        float z = pts[3 * i + 2];
        sq[i] = x * x + y * y + z * z;
    }
}

// ---------------------------------------------------------------------------
// Kernel 2: WMMA Gram tiles -> d2 chunks in LDS -> exact top-33 per row
// Block = 256 threads (8 wave32), handles 16 rows of the distance matrix.
// Chunk points/norms are staged through LDS so the WMMA inner loop is
// branch-free (no EXEC-mask divergence around v_wmma).
// ---------------------------------------------------------------------------
__global__ __launch_bounds__(256) void topk_kernel(const float* __restrict__ pts,
                                                   const float* __restrict__ sq,
                                                   int* __restrict__ idx_out) {
    __shared__ float dist[16 * CHUNK];      // 32 KB distance chunk
    __shared__ float ptsCh[CHUNK * 3];      // 6 KB: this chunk's points
    __shared__ float sqCh[CHUNK];           // 2 KB: this chunk's sq norms
    __shared__ float redV[256];
    __shared__ int   redI[256];
    __shared__ float topV[16][TOPK];
    __shared__ int   topI[16][TOPK];
    __shared__ float chV[16][TOPK];
    __shared__ int   chI[16][TOPK];
    __shared__ float tmpV[16][TOPK];
    __shared__ int   tmpI[16][TOPK];

    const int tid       = threadIdx.x;
    const int lane      = tid & 31;
    const int wid       = tid >> 5;       // wave id 0..7
    const int r0        = blockIdx.x * 16;
    const int colInLane = lane & 15;      // N (or M) position within 16
    const int hi        = lane >> 4;      // 0: lanes 0-15, 1: lanes 16-31

    // --- A-matrix (16x4 f32), branch-free build:
    //     VGPR0 = K0 (lanes 0-15) | K2 (lanes 16-31); VGPR1 = K1 | K3(pad 0)
    v2f a;
    {
        int i = r0 + colInLane;
        float ax = pts[3 * i + 2 * hi];       // x (lo lanes) or z (hi lanes)
        float ay = pts[3 * i + 1];            // y, selected away on hi lanes
        a.x = ax;
        a.y = (hi == 0) ? ay : 0.0f;          // v_cndmask, no divergence
    }
    // squared norms of this block's rows: lane needs M = k + 8*hi
    float sqA[8];
#pragma unroll
    for (int k = 0; k < 8; ++k) sqA[k] = sq[r0 + k + 8 * hi];

    if (tid < 16) {
        for (int k = 0; k < TOPK; ++k) { topV[tid][k] = __builtin_inff(); topI[tid][k] = 0; }
    }
    __syncthreads();

    for (int ch = 0; ch < NCHUNKS; ++ch) {
        const int jbase = ch * CHUNK;

        // ---- stage chunk points + norms into LDS (coalesced) ----
#pragma unroll
        for (int t = 0; t < (CHUNK * 3) / 256; ++t)       // 6 loads/thread
            ptsCh[tid + t * 256] = pts[jbase * 3 + tid + t * 256];
#pragma unroll
        for (int t = 0; t < CHUNK / 256; ++t)             // 2 loads/thread
            sqCh[tid + t * 256] = sq[jbase + tid + t * 256];
        __syncthreads();

        // ---- fill 16 x CHUNK distances via WMMA (4 tiles per wave) ----
#pragma unroll
        for (int t = 0; t < NTILES_PER_CHUNK / 8; ++t) {
            const int tt = wid * (NTILES_PER_CHUNK / 8) + t;
            const int jl = tt * 16 + colInLane;           // local column
            // branch-free B operand from LDS
            float bx = ptsCh[3 * jl + 2 * hi];
            float by = ptsCh[3 * jl + 1];
            v2f b;
            b.x = bx;
            b.y = (hi == 0) ? by : 0.0f;
            v8f c = {};
            // D = A(16x4) x B(4x16) + 0 : full-precision f32 Gram tile
            c = __builtin_amdgcn_wmma_f32_16x16x4_f32(
                    false, a, false, b, (short)0, c, false, false);
            float sqj = sqCh[jl];
#pragma unroll
            for (int k = 0; k < 8; ++k) {
                int M = k + 8 * hi;                       // C/D layout row
                float d2 = sqA[k] + sqj - 2.0f * c[k];
                dist[M * CHUNK + tt * 16 + colInLane] = d2;
            }
        }
        __syncthreads();

        // ---- exact top-33 of this chunk: 16 scanners per row ----
        const int row = tid >> 4;   // 0..15
        const int sid = tid & 15;   // scanner id
        for (int it = 0; it < TOPK; ++it) {
            float bv = __builtin_inff();
            int   bc = 0;
            for (int t = 0; t < CHUNK / 16; ++t) {
                int cc = sid + (t << 4);
                float v = dist[row * CHUNK + cc];
                if (v < bv) { bv = v; bc = cc; }
            }
            redV[tid] = bv; redI[tid] = bc;
            __syncthreads();
            if (sid == 0) {
                float mv = redV[row << 4];
                int   mc = redI[row << 4];
                for (int s = 1; s < 16; ++s) {
                    float v = redV[(row << 4) + s];
                    int   cc = redI[(row << 4) + s];
                    if (v < mv || (v == mv && cc < mc)) { mv = v; mc = cc; }
                }
                chV[row][it] = mv;
                chI[row][it] = jbase + mc;
                dist[row * CHUNK + mc] = __builtin_inff();   // knock out winner
            }
            __syncthreads();
        }

        // ---- merge chunk list into running top-33 (both sorted ascending) ----
        if (tid < 16) {
            int p = 0, q = 0;
            for (int k = 0; k < TOPK; ++k) {     // p+q = k <= 32, no overrun
                float tv = topV[tid][p], cv = chV[tid][q];
                bool takeTop = (tv < cv) || (tv == cv && topI[tid][p] <= chI[tid][q]);
                if (takeTop) { tmpV[tid][k] = tv; tmpI[tid][k] = topI[tid][p]; ++p; }
                else         { tmpV[tid][k] = cv; tmpI[tid][k] = chI[tid][q]; ++q; }
            }
            for (int k = 0; k < TOPK; ++k) { topV[tid][k] = tmpV[tid][k]; topI[tid][k] = tmpI[tid][k]; }
        }
        __syncthreads();
    }

    if (tid < 16) {
        for (int k = 0; k < TOPK; ++k)
            idx_out[(r0 + tid) * TOPK + k] = topI[tid][k];
    }
}

// ---------------------------------------------------------------------------
// Kernel 3: stream the 478 MB of outputs; feats (8 MB) is L2-resident so the
// gathered reads hit L2; writes are the bandwidth floor (~20 us @ 23.3 TB/s).
// One block per point; float4 (16B) transfers.
// ---------------------------------------------------------------------------
__global__ __launch_bounds__(256) void gather_kernel(const float* __restrict__ feats,
                                                     const int* __restrict__ idx,
                                                     float* __restrict__ out) {
    const int i   = blockIdx.x;
    const int tid = threadIdx.x;
    const float4* f4 = (const float4*)feats;
    float4* o1 = (float4*)(out + (size_t)N_PTS * 256);                  // scale 8
    float4* o2 = (float4*)(out + (size_t)N_PTS * 256 * (1 + 8));        // scale 16
    float4* o3 = (float4*)(out + (size_t)N_PTS * 256 * (1 + 8 + 16));   // scale 32

    // out0: passthrough copy of feats row i (256 floats)
    out[(size_t)i * 256 + tid] = feats[(size_t)i * 256 + tid];

    const int l4  = tid & 63;   // float4 lane within a 256-float row
    const int sub = tid >> 6;   // 0..3 neighbor sub-slot
#pragma unroll
    for (int g = 0; g < 8; ++g) {
        int k  = g * 4 + sub;                      // neighbor 0..31
        int nb = idx[i * TOPK + 1 + k];            // skip self (slot 0)
        float4 v = f4[(size_t)nb * 64 + l4];
        o3[((size_t)i * 32 + k) * 64 + l4] = v;
        if (k < 16) o2[((size_t)i * 16 + k) * 64 + l4] = v;
        if (k < 8)  o1[((size_t)i * 8  + k) * 64 + l4] = v;
    }
}

// ---------------------------------------------------------------------------
extern "C" void kernel_launch(void* const* d_in, const int* in_sizes, int n_in,
                              void* d_out, int out_size, void* d_ws, size_t ws_size,
                              hipStream_t stream) {
    const float* feats = (const float*)d_in[0];   // (8192, 256) f32
    const float* pts   = (const float*)d_in[1];   // (8192, 3)   f32
    (void)in_sizes; (void)n_in; (void)out_size; (void)ws_size;

    float* sq  = (float*)d_ws;                                        // 32 KB
    int*   idx = (int*)((char*)d_ws + (size_t)N_PTS * sizeof(float)); // ~1 MB

    sq_kernel<<<N_PTS / 256, 256, 0, stream>>>(pts, sq);
    topk_kernel<<<N_PTS / 16, 256, 0, stream>>>(pts, sq, idx);
    gather_kernel<<<N_PTS, 256, 0, stream>>>(feats, idx, (float*)d_out);
}